// Blender_38809324486930
// MI455X (gfx1250) — compile-verified
//
#include <hip/hip_runtime.h>
#include <hip/hip_bf16.h>
#include <math.h>

// Problem constants (from reference)
#define NB   4      // batch
#define NP   512    // proposals
#define NK   4      // bases / channels
#define FH   200
#define FW   336
#define AA   14     // attention patch size
#define PS   56     // pooler size
#define SCALE 0.25f

typedef __attribute__((ext_vector_type(2))) float v2f;
typedef __attribute__((ext_vector_type(8))) float v8f;

// ---- LDS layout (in floats). Column-strided A-operands padded to 17 to avoid
// bank conflicts (64 banks, stride 17 is coprime with 64).
constexpr int OFF_LP = 0;                     // Lp  [64][17]  interp matrix (padded 56x14)
constexpr int OFF_LT = OFF_LP + 64 * 17;      // LTp [16][64]  transpose (padded 14x56)
constexpr int OFF_AP = OFF_LT + 16 * 64;      // Ap  [4][16][16] attention patch (padded)
constexpr int OFF_T  = OFF_AP + 4 * 16 * 16;  // T   [4][64][17] intermediate L@A
constexpr int OFF_R  = OFF_T + 4 * 64 * 17;   // R   [4][56][57] resized attention
constexpr int LDS_TOTAL = OFF_R + 4 * 56 * 57;
constexpr int ZERO_FLOATS = OFF_T;            // Lp/LTp/Ap must be zero-initialized

__global__ __launch_bounds__(256) void blender_kernel(
    const float* __restrict__ bases,   // [NB][NK][FH][FW]
    const float* __restrict__ boxes,   // [NB][NP][4]
    const float* __restrict__ attn,    // [NB][NP][NK][AA][AA]
    float* __restrict__ out)           // [NB][NP][PS][PS]
{
    extern __shared__ float lds[];
    float (*Lp)[17]      = reinterpret_cast<float(*)[17]>(lds + OFF_LP);
    float (*LTp)[64]     = reinterpret_cast<float(*)[64]>(lds + OFF_LT);
    float (*Ap)[16][16]  = reinterpret_cast<float(*)[16][16]>(lds + OFF_AP);
    float (*T)[64][17]   = reinterpret_cast<float(*)[64][17]>(lds + OFF_T);
    float (*R)[56][57]   = reinterpret_cast<float(*)[56][57]>(lds + OFF_R);

    const int bp   = blockIdx.x;        // b*NP + p
    const int b    = bp / NP;
    const int tid  = threadIdx.x;
    const int lane = tid & 31;
    const int wid  = tid >> 5;

    const float* attn_blk = attn + (size_t)bp * (NK * AA * AA);
    const float* base_b   = bases + (size_t)b * (NK * FH * FW);

    // gfx1250 prefetch (global_prefetch_b8) of this block's attention patch
    __builtin_prefetch(attn_blk, 0, 0);

    // ---- Phase 0: zero the padded staging areas ----
    for (int i = tid; i < ZERO_FLOATS; i += 256) lds[i] = 0.0f;
    __syncthreads();

    // ---- Phase 1a: build align-corners interpolation matrix L[56][14] ----
    if (tid < PS) {
        const int o = tid;
        const float src = (float)o * (13.0f / 55.0f);   // (A-1)/(PS-1)
        int i0 = (int)floorf(src);
        i0 = i0 < 0 ? 0 : (i0 > AA - 2 ? AA - 2 : i0);
        const float t = src - (float)i0;
        Lp[o][i0]      = 1.0f - t;
        Lp[o][i0 + 1]  = t;
        LTp[i0][o]     = 1.0f - t;
        LTp[i0 + 1][o] = t;
    }
    // ---- Phase 1b: stage attention [4][14][14] into padded [4][16][16] ----
    for (int i = tid; i < NK * AA * AA; i += 256) {
        const int c = i / (AA * AA);
        const int r = (i / AA) % AA;
        const int q = i % AA;
        Ap[c][r][q] = attn_blk[i];
    }
    __syncthreads();

    // Shared WMMA lane mapping (32-bit operands, wave32):
    //   A 16x4: m = lane&15, k = 2*(lane>>4) + v
    //   B 4x16: n = lane&15, k = 2*(lane>>4) + v
    //   C/D:    N = lane&15, M = r + 8*(lane>>4)
    const int khalf = (lane >> 4) << 1;   // 0 or 2
    const int lrow  = lane & 15;
    const int mhalf = (lane >> 4) << 3;   // 0 or 8

    // ---- Phase 2: GEMM1  T[c] (64x16) = Lp (64x16) @ Ap[c] (16x16) ----
    // 16 output tiles over 8 waves -> exactly 2 per wave (compile-time count:
    // no exec-mask loop; WMMA always runs with EXEC == all-1s by construction).
    #pragma unroll
    for (int it = 0; it < 2; ++it) {
        const int w  = it * 8 + wid;      // 0..15
        const int c  = w >> 2;
        const int mt = w & 3;
        v8f acc = {};
        #pragma unroll
        for (int ks = 0; ks < 4; ++ks) {              // K = 16 in steps of 4
            const int kk = ks * 4 + khalf;
            v2f a, bm;
            a[0]  = Lp[mt * 16 + lrow][kk];
            a[1]  = Lp[mt * 16 + lrow][kk + 1];
            bm[0] = Ap[c][kk][lrow];
            bm[1] = Ap[c][kk + 1][lrow];
            acc = __builtin_amdgcn_wmma_f32_16x16x4_f32(
                      false, a, false, bm, (short)0, acc, false, false);
        }
        #pragma unroll
        for (int r = 0; r < 8; ++r)
            T[c][mt * 16 + mhalf + r][lrow] = acc[r];
    }
    __syncthreads();

    // ---- Phase 3: GEMM2  R[c] (64x64) = T[c] (64x16) @ LTp (16x64) ----
    // 64 output tiles over 8 waves -> exactly 8 per wave. Unroll 2 tiles for
    // independent accumulators (hides WMMA latency) without VGPR spills.
    #pragma unroll 2
    for (int it = 0; it < 8; ++it) {
        const int w  = it * 8 + wid;      // 0..63
        const int c  = w >> 4;
        const int mt = (w >> 2) & 3;
        const int nt = w & 3;
        v8f acc = {};
        #pragma unroll
        for (int ks = 0; ks < 4; ++ks) {
            const int kk = ks * 4 + khalf;
            v2f a, bm;
            a[0]  = T[c][mt * 16 + lrow][kk];
            a[1]  = T[c][mt * 16 + lrow][kk + 1];
            bm[0] = LTp[kk][nt * 16 + lrow];
            bm[1] = LTp[kk + 1][nt * 16 + lrow];
            acc = __builtin_amdgcn_wmma_f32_16x16x4_f32(
                      false, a, false, bm, (short)0, acc, false, false);
        }
        const int n = nt * 16 + lrow;
        if (n < PS) {
            #pragma unroll
            for (int r = 0; r < 8; ++r) {
                const int m = mt * 16 + mhalf + r;
                if (m < PS) R[c][m][n] = acc[r];
            }
        }
    }
    __syncthreads();

    // ---- Phase 4: per-pixel softmax over K + RoI-align gather + blend ----
    const float* bx = boxes + (size_t)bp * 4;
    const float x1 = bx[0] * SCALE;
    const float y1 = bx[1] * SCALE;
    const float x2 = bx[2] * SCALE;
    const float y2 = bx[3] * SCALE;
    const float rw = fmaxf(x2 - x1, 1.0f);
    const float rh = fmaxf(y2 - y1, 1.0f);
    const float bw = rw * (1.0f / (float)PS);
    const float bh = rh * (1.0f / (float)PS);

    float* out_blk = out + (size_t)bp * (PS * PS);

    for (int idx = tid; idx < PS * PS; idx += 256) {
        const int py = idx / PS;
        const int px = idx - py * PS;

        // softmax over the 4 channels of resized attention
        const float a0 = R[0][py][px];
        const float a1 = R[1][py][px];
        const float a2 = R[2][py][px];
        const float a3 = R[3][py][px];
        const float mx = fmaxf(fmaxf(a0, a1), fmaxf(a2, a3));
        float e[NK];
        e[0] = __expf(a0 - mx);
        e[1] = __expf(a1 - mx);
        e[2] = __expf(a2 - mx);
        e[3] = __expf(a3 - mx);
        const float inv = 1.0f / (e[0] + e[1] + e[2] + e[3]);

        // RoI-align sample position (sampling_ratio=1 -> bin center)
        const float sx = x1 + ((float)px + 0.5f) * bw;
        const float sy = y1 + ((float)py + 0.5f) * bh;
        const bool valid = (sy > -1.0f) && (sy < (float)FH) &&
                           (sx > -1.0f) && (sx < (float)FW);
        const float yc = fminf(fmaxf(sy, 0.0f), (float)(FH - 1));
        const float xc = fminf(fmaxf(sx, 0.0f), (float)(FW - 1));
        const int y0 = (int)floorf(yc);
        const int x0 = (int)floorf(xc);
        const int y1i = min(y0 + 1, FH - 1);
        const int x1i = min(x0 + 1, FW - 1);
        const float ly = yc - (float)y0;
        const float lx = xc - (float)x0;
        const float w00 = (1.0f - ly) * (1.0f - lx);
        const float w01 = (1.0f - ly) * lx;
        const float w10 = ly * (1.0f - lx);
        const float w11 = ly * lx;
        const int o00 = y0 * FW + x0;
        const int o01 = y0 * FW + x1i;
        const int o10 = y1i * FW + x0;
        const int o11 = y1i * FW + x1i;
        const float vm = valid ? 1.0f : 0.0f;

        float acc = 0.0f;
        #pragma unroll
        for (int c = 0; c < NK; ++c) {
            const float* f = base_b + c * (FH * FW);
            const float v = f[o00] * w00 + f[o01] * w01 +
                            f[o10] * w10 + f[o11] * w11;
            acc += (e[c] * inv) * (v * vm);
        }
        out_blk[idx] = acc;
    }
}

extern "C" void kernel_launch(void* const* d_in, const int* in_sizes, int n_in,
                              void* d_out, int out_size, void* d_ws, size_t ws_size,
                              hipStream_t stream) {
    (void)in_sizes; (void)n_in; (void)out_size; (void)d_ws; (void)ws_size;
    const float* bases = (const float*)d_in[0];
    const float* boxes = (const float*)d_in[1];
    const float* attn  = (const float*)d_in[2];
    float* out = (float*)d_out;
    const size_t ldsBytes = (size_t)LDS_TOTAL * sizeof(float);  // ~79 KB of 320 KB/WGP
    blender_kernel<<<dim3(NB * NP), dim3(256), ldsBytes, stream>>>(bases, boxes, attn, out);
}